// OPTNET_16930761081154
// MI455X (gfx1250) — compile-verified
//
#include <hip/hip_runtime.h>
#include <hip/hip_bf16.h>
#include <math.h>

// Problem constants (match setup_inputs): B=128, m=256, n=1024
#define BATCH 128
#define M_DIM 256
#define N_DIM 1024
#define N_ITERS 200
#define JITTER 1e-6f

typedef __attribute__((ext_vector_type(2))) float v2f;
typedef __attribute__((ext_vector_type(8))) float v8f;

// ---------------------------------------------------------------------------
// Kernel 1: AAt[b] = A[b] * A[b]^T + JITTER*I   (WMMA f32 16x16x4)
// One wave computes one 16x16 output tile, K=1024 in steps of 4.
// Layouts (ISA 7.12.2):
//   A-op 16x4 : lane L -> M = L&15, K-pair = 2*(L>>4) in v[0],v[1]
//   B-op 4x16 : lane L -> N = L&15, K-pair = 2*(L>>4) in v[0],v[1]
//               (B tile = A^T tile, so B[kk][j] = A[j0+j][k0+kk])
//   C/D 16x16 : VGPR r, lane L -> row = i0 + r + 8*(L>>4), col = j0 + (L&15)
// ---------------------------------------------------------------------------
__global__ void aat_wmma_kernel(const float* __restrict__ A,
                                float* __restrict__ C) {
    const int wgPerBatch = 32;                 // 32 WGs * 8 waves = 256 tiles
    int batch = blockIdx.x / wgPerBatch;
    int wg    = blockIdx.x % wgPerBatch;
    int lane  = threadIdx.x & 31;
    int wave  = threadIdx.x >> 5;
    int tile  = wg * 8 + wave;                 // 0..255
    int i0 = (tile >> 4) * 16;                 // output row block
    int j0 = (tile & 15) * 16;                 // output col block

    const float* Ab = A + (size_t)batch * M_DIM * N_DIM;
    int Mi = lane & 15;
    int kh = lane >> 4;                        // 0/1 -> K pair {2kh, 2kh+1}
    const float* arow = Ab + (size_t)(i0 + Mi) * N_DIM + 2 * kh;
    const float* brow = Ab + (size_t)(j0 + Mi) * N_DIM + 2 * kh;

    v8f acc0 = {};
    v8f acc1 = {};
    // Two independent accumulator chains to hide WMMA D->C latency.
    #pragma unroll 4
    for (int k0 = 0; k0 < N_DIM; k0 += 8) {
        v2f a0 = *(const v2f*)(arow + k0);
        v2f b0 = *(const v2f*)(brow + k0);
        v2f a1 = *(const v2f*)(arow + k0 + 4);
        v2f b1 = *(const v2f*)(brow + k0 + 4);
        acc0 = __builtin_amdgcn_wmma_f32_16x16x4_f32(
            false, a0, false, b0, (short)0, acc0, false, false);
        acc1 = __builtin_amdgcn_wmma_f32_16x16x4_f32(
            false, a1, false, b1, (short)0, acc1, false, false);
    }

    float* Cb = C + (size_t)batch * M_DIM * M_DIM;
    int col  = j0 + Mi;
    int row0 = i0 + kh * 8;
    #pragma unroll
    for (int r = 0; r < 8; ++r) {
        float v = acc0[r] + acc1[r];
        int row = row0 + r;
        if (row == col) v += JITTER;           // AAt + eps*I
        Cb[(size_t)row * M_DIM + col] = v;
    }
}

// ---------------------------------------------------------------------------
// Kernel 2: per-batch  L = chol(AAt)  (in place),  Linv = L^-1 (full matrix,
// upper triangle zero-filled),  Minv = Linv^T * Linv via WMMA (overwrites
// the AAt/L region). One 256-thread workgroup per batch; one-time cost.
// ---------------------------------------------------------------------------
__global__ void chol_inv_kernel(float* __restrict__ Mreg,   // AAt -> L -> Minv
                                float* __restrict__ Linv) {
    int batch = blockIdx.x;
    float* L  = Mreg + (size_t)batch * M_DIM * M_DIM;
    float* Li = Linv + (size_t)batch * M_DIM * M_DIM;
    int tid  = threadIdx.x;                    // 0..255
    int lane = tid & 31;
    int wave = tid >> 5;

    // --- Cholesky (Crout, column by column, lower triangle only) ---
    for (int j = 0; j < M_DIM; ++j) {
        if (wave == 0) {                       // wave 0 reduces the diagonal
            float s = 0.f;
            for (int k = lane; k < j; k += 32) {
                float v = L[(size_t)j * M_DIM + k];
                s += v * v;
            }
            for (int off = 16; off > 0; off >>= 1) s += __shfl_down(s, off, 32);
            if (lane == 0) {
                float d = L[(size_t)j * M_DIM + j] - s;
                L[(size_t)j * M_DIM + j] = sqrtf(fmaxf(d, 1e-20f));
            }
        }
        __syncthreads();
        float d = L[(size_t)j * M_DIM + j];
        int i = j + 1 + tid;                   // one row per thread
        if (i < M_DIM) {
            float s = L[(size_t)i * M_DIM + j];
            for (int k = 0; k < j; ++k)
                s -= L[(size_t)i * M_DIM + k] * L[(size_t)j * M_DIM + k];
            L[(size_t)i * M_DIM + j] = s / d;
        }
        __syncthreads();
    }

    // --- Linv: thread j computes column j; zero the strict upper part so the
    //     subsequent full-K WMMA GEMM reads only valid data ---
    {
        int j = tid;
        for (int i = 0; i < j; ++i) Li[(size_t)i * M_DIM + j] = 0.0f;
        for (int i = j; i < M_DIM; ++i) {
            float s = (i == j) ? 1.0f : 0.0f;
            for (int k = j; k < i; ++k)
                s -= L[(size_t)i * M_DIM + k] * Li[(size_t)k * M_DIM + j];
            Li[(size_t)i * M_DIM + j] = s / L[(size_t)i * M_DIM + i];
        }
    }
    __syncthreads();

    // --- Minv = Linv^T * Linv via WMMA f32 16x16x4 ---
    // A-op[M][K] = Linv[k][a0+M] (transposed read), B-op[K][N] = Linv[k][b0+N]
    int Mi = lane & 15;
    int kh = lane >> 4;
    for (int tile = wave; tile < 256; tile += 8) {  // 256 tiles over 8 waves
        int a0 = (tile >> 4) * 16;
        int b0 = (tile & 15) * 16;
        v8f acc = {};
        for (int k0 = 0; k0 < M_DIM; k0 += 4) {
            int kr = k0 + 2 * kh;
            v2f av, bv;
            av[0] = Li[(size_t)kr * M_DIM + a0 + Mi];
            av[1] = Li[(size_t)(kr + 1) * M_DIM + a0 + Mi];
            bv[0] = Li[(size_t)kr * M_DIM + b0 + Mi];
            bv[1] = Li[(size_t)(kr + 1) * M_DIM + b0 + Mi];
            acc = __builtin_amdgcn_wmma_f32_16x16x4_f32(
                false, av, false, bv, (short)0, acc, false, false);
        }
        #pragma unroll
        for (int r = 0; r < 8; ++r)
            L[(size_t)(a0 + r + 8 * kh) * M_DIM + (b0 + Mi)] = acc[r];
    }
}

// ---------------------------------------------------------------------------
// Kernel 3: 200 Dykstra iterations, one persistent 1024-thread workgroup per
// batch (32 waves = 8 waves/SIMD32 for L2-latency hiding; A is L2-resident).
// State (x,p,q,t,b,r,z,mask) lives in LDS; A and Minv stream from L2.
//   proj_affine(v) = v - A^T Minv (A v - b)
// ---------------------------------------------------------------------------
__global__ void dykstra_kernel(const float* __restrict__ x0,
                               const float* __restrict__ bvec,
                               const float* __restrict__ A,
                               const int* __restrict__ mask,
                               const float* __restrict__ Minv,
                               float* __restrict__ out) {
    __shared__ float xs[N_DIM], ps[N_DIM], qs[N_DIM], ts[N_DIM];
    __shared__ int   ms[N_DIM];
    __shared__ float bs[M_DIM], rs[M_DIM], zs[M_DIM];

    int batch = blockIdx.x;
    int tid   = threadIdx.x;                   // 1024 threads = 32 waves
    int lane  = tid & 31;
    int wave  = tid >> 5;
    const float* Ab = A    + (size_t)batch * M_DIM * N_DIM;
    const float* Mb = Minv + (size_t)batch * M_DIM * M_DIM;

    for (int i = tid; i < N_DIM; i += 1024) {
        xs[i] = x0[(size_t)batch * N_DIM + i];
        ps[i] = 0.f;
        qs[i] = 0.f;
        ms[i] = mask[i];
    }
    if (tid < M_DIM) bs[tid] = bvec[(size_t)batch * M_DIM + tid];
    __syncthreads();

    for (int it = 0; it <= N_ITERS; ++it) {
        // t = x + p   (last pass: final affine projection of x itself)
        for (int i = tid; i < N_DIM; i += 1024)
            ts[i] = (it < N_ITERS) ? (xs[i] + ps[i]) : xs[i];
        __syncthreads();

        // r = A t - b : each wave owns 8 rows, lanes stride K (coalesced)
        for (int row = wave * 8; row < wave * 8 + 8; ++row) {
            const float* Ar = Ab + (size_t)row * N_DIM;
            __builtin_prefetch(Ar + N_DIM, 0, 1);      // next row into cache
            float acc = 0.f;
            for (int k = lane; k < N_DIM; k += 32) acc += Ar[k] * ts[k];
            for (int off = 16; off > 0; off >>= 1)
                acc += __shfl_down(acc, off, 32);
            if (lane == 0) rs[row] = acc - bs[row];
        }
        __syncthreads();

        // z = Minv r
        for (int row = wave * 8; row < wave * 8 + 8; ++row) {
            const float* Mr = Mb + (size_t)row * M_DIM;
            float acc = 0.f;
            for (int k = lane; k < M_DIM; k += 32) acc += Mr[k] * rs[k];
            for (int off = 16; off > 0; off >>= 1)
                acc += __shfl_down(acc, off, 32);
            if (lane == 0) zs[row] = acc;
        }
        __syncthreads();

        // y = t - A^T z ; one column per thread, reads coalesced across tid
        for (int c = tid; c < N_DIM; c += 1024) {
            float acc = 0.f;
            #pragma unroll 4
            for (int k = 0; k < M_DIM; ++k)
                acc += Ab[(size_t)k * N_DIM + c] * zs[k];
            float y = ts[c] - acc;
            if (it < N_ITERS) {
                // Dykstra update
                float pnew = ts[c] - y;               // = acc
                float v    = y + qs[c];
                float xn   = ms[c] ? fmaxf(v, 0.f) : v;
                ps[c] = pnew;
                qs[c] = v - xn;
                xs[c] = xn;
            } else {
                out[(size_t)batch * N_DIM + c] = y;   // final projection
            }
        }
        __syncthreads();
    }
}

// ---------------------------------------------------------------------------
extern "C" void kernel_launch(void* const* d_in, const int* in_sizes, int n_in,
                              void* d_out, int out_size, void* d_ws, size_t ws_size,
                              hipStream_t stream) {
    const float* x0  = (const float*)d_in[0];   // [B, n]
    const float* b   = (const float*)d_in[1];   // [B, m]
    const float* A   = (const float*)d_in[2];   // [B, m, n]
    const int* mask  = (const int*)d_in[3];     // [n]
    float* out       = (float*)d_out;           // [B, n]

    // Workspace: region1 = AAt -> L -> Minv (32 MB), region2 = Linv (32 MB)
    float* region1 = (float*)d_ws;
    float* region2 = region1 + (size_t)BATCH * M_DIM * M_DIM;

    aat_wmma_kernel<<<dim3(BATCH * 32), dim3(256), 0, stream>>>(A, region1);
    chol_inv_kernel<<<dim3(BATCH), dim3(256), 0, stream>>>(region1, region2);
    dykstra_kernel<<<dim3(BATCH), dim3(1024), 0, stream>>>(x0, b, A, mask,
                                                           region1, out);
    (void)in_sizes; (void)n_in; (void)out_size; (void)ws_size;
}